// SimpleAttention_62586263437810
// MI455X (gfx1250) — compile-verified
//
#include <hip/hip_runtime.h>
#include <math.h>

typedef __attribute__((ext_vector_type(2))) float v2f;
typedef __attribute__((ext_vector_type(8))) float v8f;

#define N_TOK 8192
#define DIM   1024
#define BM 128
#define BN 128
#define BK 32
#define LDSS 34   // padded LDS row stride (floats): conflict-free frag reads

// ---- CDNA5 async global->LDS staging (ASYNCcnt path) ---------------------
// Probe-confirmed prototypes (from hipcc diagnostics):
//   b128: (int __vector(4) AS1*, int __vector(4) AS3*, imm int, imm int)
//   b32 : (int AS1*,            int AS3*,            imm int, imm int)
#if __has_builtin(__builtin_amdgcn_global_load_async_to_lds_b128) && \
    __has_builtin(__builtin_amdgcn_global_load_async_to_lds_b32)  && \
    __has_builtin(__builtin_amdgcn_s_wait_asynccnt)
#define HAVE_ASYNC 1
#define AS1 __attribute__((address_space(1)))
#define AS3 __attribute__((address_space(3)))
typedef int gv4i __attribute__((vector_size(4 * sizeof(int))));
#define ASYNC_CP_B128(lds, gp) \
    __builtin_amdgcn_global_load_async_to_lds_b128((AS1 gv4i*)(gp), (AS3 gv4i*)(lds), 0, 0)
#define ASYNC_CP_B32(lds, gp) \
    __builtin_amdgcn_global_load_async_to_lds_b32((AS1 int*)(gp), (AS3 int*)(lds), 0, 0)
#define ASYNC_WAIT() __builtin_amdgcn_s_wait_asynccnt(0)
#else
#define HAVE_ASYNC 0
#define ASYNC_WAIT()
#endif

// ---------------------------------------------------------------------------
// Kernel 1: S = X @ X^T, diagonal forced to -inf.  Raw logits -> attn buffer.
// fp32 WMMA (V_WMMA_F32_16X16X4_F32), 128x128 tile, BK=32, 8 waves,
// double-buffered LDS with async global->LDS staging.
// ---------------------------------------------------------------------------
__global__ __launch_bounds__(256)
void attn_logits_kernel(const float* __restrict__ X, float* __restrict__ S) {
    __shared__ float As[2][BM * LDSS];
    __shared__ float Bs[2][BN * LDSS];

    const int tid  = threadIdx.x;
    const int lane = tid & 31;
    const int wave = tid >> 5;
    const int iBase = blockIdx.y * BM;
    const int jBase = blockIdx.x * BN;

    const int r  = lane & 15;   // A-frag M / B-frag N within 16-tile
    const int hi = lane >> 4;   // K-half select (ISA 16x4 f32 layout)

    v8f acc[8] = {};            // 16x128 strip per wave

    const int lrow = tid >> 3;        // 32 rows/pass, 8 threads/row
    const int lcol = (tid & 7) * 4;   // float4 column

    auto stage = [&](int buf, int k0) {
        #pragma unroll
        for (int p = 0; p < 4; ++p) {
            const int row = lrow + p * 32;
            const float* ga = X + (size_t)(iBase + row) * DIM + k0 + lcol;
            const float* gb = X + (size_t)(jBase + row) * DIM + k0 + lcol;
            float* la = &As[buf][row * LDSS + lcol];
            float* lb = &Bs[buf][row * LDSS + lcol];
#if HAVE_ASYNC
            ASYNC_CP_B128(la, ga);
            ASYNC_CP_B128(lb, gb);
#else
            const float4 a = *(const float4*)ga;
            la[0] = a.x; la[1] = a.y; la[2] = a.z; la[3] = a.w;
            const float4 b = *(const float4*)gb;
            lb[0] = b.x; lb[1] = b.y; lb[2] = b.z; lb[3] = b.w;
#endif
        }
    };

    stage(0, 0);
    ASYNC_WAIT();
    __syncthreads();

    const int NT = DIM / BK;
    const int mOff = wave * 16;
    for (int t = 0; t < NT; ++t) {
        const int cur = t & 1;
        if (t + 1 < NT) stage(cur ^ 1, (t + 1) * BK);

        #pragma unroll
        for (int kk = 0; kk < BK; kk += 4) {
            const v2f a = *(const v2f*)&As[cur][(mOff + r) * LDSS + kk + 2 * hi];
            #pragma unroll
            for (int s = 0; s < 8; ++s) {
                const v2f b = *(const v2f*)&Bs[cur][(s * 16 + r) * LDSS + kk + 2 * hi];
                acc[s] = __builtin_amdgcn_wmma_f32_16x16x4_f32(
                    false, a, false, b, (short)0, acc[s], false, false);
            }
        }
        if (t + 1 < NT) ASYNC_WAIT();
        __syncthreads();   // publishes next tile; retires reads of buffer cur
    }

    // Diagonal mask + store. C/D layout: VGPR v -> M = v + 8*hi, N = r.
    const int gRow0 = iBase + wave * 16 + hi * 8;
    #pragma unroll
    for (int s = 0; s < 8; ++s) {
        const int gCol = jBase + s * 16 + r;
        #pragma unroll
        for (int v = 0; v < 8; ++v) {
            const int gRow = gRow0 + v;
            float val = acc[s][v];
            if (gRow == gCol) val = -INFINITY;
            S[(size_t)gRow * N_TOK + gCol] = val;
        }
    }
}

// ---------------------------------------------------------------------------
// Kernel 2: row-wise softmax in place over 8192 columns. One 256-thread block
// per row; 32 values/thread in registers -> single read + single write.
// ---------------------------------------------------------------------------
__global__ __launch_bounds__(256)
void softmax_kernel(float* __restrict__ S) {
    float* rp = S + (size_t)blockIdx.x * N_TOK;
    const int tid = threadIdx.x;

    float vals[32];
    float m = -INFINITY;
    #pragma unroll
    for (int i = 0; i < 32; ++i) {
        vals[i] = rp[tid + i * 256];
        m = fmaxf(m, vals[i]);
    }
    #pragma unroll
    for (int off = 16; off > 0; off >>= 1)
        m = fmaxf(m, __shfl_xor(m, off, 32));

    __shared__ float red[8];
    if ((tid & 31) == 0) red[tid >> 5] = m;
    __syncthreads();
    m = red[0];
    #pragma unroll
    for (int w = 1; w < 8; ++w) m = fmaxf(m, red[w]);
    __syncthreads();

    float sum = 0.0f;
    #pragma unroll
    for (int i = 0; i < 32; ++i) {
        vals[i] = __expf(vals[i] - m);   // diag: exp(-inf - m) == 0
        sum += vals[i];
    }
    #pragma unroll
    for (int off = 16; off > 0; off >>= 1)
        sum += __shfl_xor(sum, off, 32);
    if ((tid & 31) == 0) red[tid >> 5] = sum;
    __syncthreads();
    sum = 0.0f;
    #pragma unroll
    for (int w = 0; w < 8; ++w) sum += red[w];

    const float inv = 1.0f / sum;
    #pragma unroll
    for (int i = 0; i < 32; ++i)
        rp[tid + i * 256] = vals[i] * inv;
}

// ---------------------------------------------------------------------------
// Kernel 3: out = P @ X  (8192x8192 @ 8192x1024). Same WMMA structure;
// B tile (X) staged transposed in LDS; double-buffered async staging.
// ---------------------------------------------------------------------------
__global__ __launch_bounds__(256)
void attn_out_kernel(const float* __restrict__ P, const float* __restrict__ X,
                     float* __restrict__ O) {
    __shared__ float As[2][BM * LDSS];   // P rows   [BM][BK]
    __shared__ float Bt[2][BN * LDSS];   // X^T tile [BN][BK]

    const int tid  = threadIdx.x;
    const int lane = tid & 31;
    const int wave = tid >> 5;
    const int iBase = blockIdx.y * BM;   // rows of P / out
    const int nBase = blockIdx.x * BN;   // cols of X / out

    const int r  = lane & 15;
    const int hi = lane >> 4;

    v8f acc[8] = {};

    const int lrow = tid >> 3;
    const int lcol = (tid & 7) * 4;
    const int bn   = tid & 127;      // column of X handled by this thread
    const int bk0  = tid >> 7;       // starting k within tile (0 or 1)

    auto stage = [&](int buf, int k0) {
        #pragma unroll
        for (int p = 0; p < 4; ++p) {
            const int row = lrow + p * 32;
            const float* ga = P + (size_t)(iBase + row) * N_TOK + k0 + lcol;
            float* la = &As[buf][row * LDSS + lcol];
#if HAVE_ASYNC
            ASYNC_CP_B128(la, ga);
#else
            const float4 a = *(const float4*)ga;
            la[0] = a.x; la[1] = a.y; la[2] = a.z; la[3] = a.w;
#endif
        }
        // Transposed X tile: Bt[n][kk] = X[k0+kk][nBase+n]
        // (coalesced along n; stride-34 scatter is conflict-free for b32)
        #pragma unroll
        for (int p = 0; p < 16; ++p) {
            const int kk = bk0 + p * 2;
            const float* gb = X + (size_t)(k0 + kk) * DIM + nBase + bn;
            float* lb = &Bt[buf][bn * LDSS + kk];
#if HAVE_ASYNC
            ASYNC_CP_B32(lb, gb);
#else
            *lb = *gb;
#endif
        }
    };

    stage(0, 0);
    ASYNC_WAIT();
    __syncthreads();

    const int NT = N_TOK / BK;
    const int mOff = wave * 16;
    for (int t = 0; t < NT; ++t) {
        const int cur = t & 1;
        if (t + 1 < NT) stage(cur ^ 1, (t + 1) * BK);

        #pragma unroll
        for (int kk = 0; kk < BK; kk += 4) {
            const v2f a = *(const v2f*)&As[cur][(mOff + r) * LDSS + kk + 2 * hi];
            #pragma unroll
            for (int s = 0; s < 8; ++s) {
                const v2f b = *(const v2f*)&Bt[cur][(s * 16 + r) * LDSS + kk + 2 * hi];
                acc[s] = __builtin_amdgcn_wmma_f32_16x16x4_f32(
                    false, a, false, b, (short)0, acc[s], false, false);
            }
        }
        if (t + 1 < NT) ASYNC_WAIT();
        __syncthreads();
    }

    const int gRow0 = iBase + wave * 16 + hi * 8;
    #pragma unroll
    for (int s = 0; s < 8; ++s) {
        const int gCol = nBase + s * 16 + r;
        #pragma unroll
        for (int v = 0; v < 8; ++v)
            O[(size_t)(gRow0 + v) * DIM + gCol] = acc[s][v];
    }
}

// ---------------------------------------------------------------------------
extern "C" void kernel_launch(void* const* d_in, const int* in_sizes, int n_in,
                              void* d_out, int out_size, void* d_ws, size_t ws_size,
                              hipStream_t stream) {
    (void)in_sizes; (void)n_in; (void)out_size; (void)d_ws; (void)ws_size;

    const float* X = (const float*)d_in[0];
    float* out  = (float*)d_out;                  // [8192][1024]
    float* attn = out + (size_t)N_TOK * DIM;      // [8192][8192]

    attn_logits_kernel<<<dim3(N_TOK / BN, N_TOK / BM), 256, 0, stream>>>(X, attn);
    softmax_kernel<<<N_TOK, 256, 0, stream>>>(attn);
    attn_out_kernel<<<dim3(DIM / BN, N_TOK / BM), 256, 0, stream>>>(attn, X, out);
}